// OnlyLinear_2_10728828305957
// MI455X (gfx1250) — compile-verified
//
#include <hip/hip_runtime.h>

typedef __attribute__((ext_vector_type(2))) float v2f;
typedef __attribute__((ext_vector_type(8))) float v8f;

// ---------------------------------------------------------------------------
// Kernel 1: fold the 9-layer affine chain into W_eff (10x784, padded to
// 16x784) and b_eff (10, padded to 16). All math in fp64; sizes are tiny
// (worst stage: 16*784 outputs x 69-MAC dots), one block is plenty.
// ---------------------------------------------------------------------------
struct FoldParams {
  const float* W[9];
  const float* b[9];
  float* Weff;   // [16][784] in d_ws, rows 10..15 zero
  float* beff;   // [16] in d_ws, entries 10..15 zero
};

__global__ __launch_bounds__(256) void fold_kernel(FoldParams p) {
  // layer dims: Wi has shape (d[i+1], d[i]) row-major
  constexpr int d[10] = {784, 69, 31, 10, 10, 10, 10, 10, 10, 10};
  __shared__ double S[2][10 * 69];   // running product W8*...*Wi  (<= 10x69)
  __shared__ double bb[2][69];       // running folded bias        (<= 69)
  const int tid = threadIdx.x;
  const int nt  = blockDim.x;

  // S = W8 (10x10), bb = b0 (69)
  for (int i = tid; i < d[9] * d[8]; i += nt) S[0][i] = (double)p.W[8][i];
  for (int i = tid; i < d[1]; i += nt)        bb[0][i] = (double)p.b[0][i];
  __syncthreads();

  // S <- S @ W_L  for L = 7 .. 1   (S: 10 x d[L+1], W_L: d[L+1] x d[L])
  int cs = 0;
  for (int L = 7; L >= 1; --L) {
    const int K = d[L + 1], C = d[L];
    for (int i = tid; i < 10 * C; i += nt) {
      const int r = i / C, c = i % C;
      double s = 0.0;
      for (int k = 0; k < K; ++k)
        s += S[cs][r * K + k] * (double)p.W[L][k * C + c];
      S[cs ^ 1][r * C + c] = s;
    }
    __syncthreads();
    cs ^= 1;
  }

  // bb <- W_L @ bb + b_L  for L = 1 .. 8
  int cb = 0;
  for (int L = 1; L <= 8; ++L) {
    const int In = d[L], On = d[L + 1];
    for (int i = tid; i < On; i += nt) {
      double s = (double)p.b[L][i];
      for (int k = 0; k < In; ++k)
        s += (double)p.W[L][i * In + k] * bb[cb][k];
      bb[cb ^ 1][i] = s;
    }
    __syncthreads();
    cb ^= 1;
  }

  // W_eff = S (10x69) @ W0 (69x784), zero-padded to 16 rows
  for (int i = tid; i < 16 * 784; i += nt) {
    const int r = i / 784, c = i % 784;
    double s = 0.0;
    if (r < 10)
      for (int k = 0; k < 69; ++k)
        s += S[cs][r * 69 + k] * (double)p.W[0][k * 784 + c];
    p.Weff[i] = (float)s;
  }
  for (int i = tid; i < 16; i += nt)
    p.beff[i] = (i < 10) ? (float)bb[cb][i] : 0.0f;
}

// ---------------------------------------------------------------------------
// Kernel 2: out[65536,10] = x[65536,784] @ W_eff^T + b_eff
// fp32 WMMA (V_WMMA_F32_16X16X4_F32), one 16-row tile per wave32.
// A layout (16x4): lane l -> row (l&15), K-offset 2*(l>>4)  => 8B global load
// B layout (4x16): lane l -> col (l&15), K-offset 2*(l>>4)  => 8B LDS load
// C/D (16x16 f32, 8 VGPRs): VGPR v -> row v + 8*(l>>4), col (l&15)
// ---------------------------------------------------------------------------
__global__ __launch_bounds__(256) void mlp_gemm_kernel(
    const float* __restrict__ x, const float* __restrict__ Weff,
    const float* __restrict__ beff, float* __restrict__ out) {
  __shared__ float sW[16 * 784];   // 50176 B: padded W_eff staged once/block
  __shared__ float sb[16];

  const int tid = threadIdx.x;
  for (int i = tid; i < 16 * 784; i += 256) sW[i] = Weff[i];
  if (tid < 16) sb[tid] = beff[tid];
  __syncthreads();

  const int wave = tid >> 5;
  const int lane = tid & 31;
  const int m    = lane & 15;       // A: M row / B: N col owned by this lane
  const int half = lane >> 4;       // 0: K=0,1  1: K=2,3
  const int ko   = half * 2;
  const long rowbase = (long)blockIdx.x * 128 + (long)wave * 16;

  const float* ap = x + (rowbase + m) * 784 + ko;   // 8B-aligned
  const float* bp = sW + m * 784 + ko;              // 8B-aligned (LDS)

  v8f acc = {};
#pragma unroll 4
  for (int kk = 0; kk < 196; ++kk) {
    v2f a = *(const v2f*)(ap + kk * 4);
    v2f b = *(const v2f*)(bp + kk * 4);
    // (neg_a, A, neg_b, B, c_mod, C, reuse_a, reuse_b)
    acc = __builtin_amdgcn_wmma_f32_16x16x4_f32(
        false, a, false, b, (short)0, acc, false, false);
  }

  const float bias = sb[m];         // every acc element sits in column m
#pragma unroll
  for (int v = 0; v < 8; ++v) acc[v] += bias;

  if (m < 10) {                     // only real output columns
    const long row0 = rowbase + half * 8;
#pragma unroll
    for (int v = 0; v < 8; ++v) out[(row0 + v) * 10 + m] = acc[v];
  }
}

// ---------------------------------------------------------------------------
extern "C" void kernel_launch(void* const* d_in, const int* in_sizes, int n_in,
                              void* d_out, int out_size, void* d_ws,
                              size_t ws_size, hipStream_t stream) {
  (void)in_sizes; (void)n_in; (void)out_size; (void)ws_size;
  FoldParams p;
  const float* x = (const float*)d_in[0];
  for (int i = 0; i < 9; ++i) {
    p.W[i] = (const float*)d_in[1 + 2 * i];
    p.b[i] = (const float*)d_in[2 + 2 * i];
  }
  p.Weff = (float*)d_ws;            // 16*784 floats
  p.beff = p.Weff + 16 * 784;       // 16 floats  (~50.2 KB total scratch)

  fold_kernel<<<1, 256, 0, stream>>>(p);
  mlp_gemm_kernel<<<512, 256, 0, stream>>>(x, p.Weff, p.beff, (float*)d_out);
}